// SODLoss_18047452578490
// MI455X (gfx1250) — compile-verified
//
#include <hip/hip_runtime.h>

// SOD structure loss, (64,1,512,512) fp32 -> scalar.
// HBM-bound: 134 MB compulsory -> ~5.8us floor @ 23.3 TB/s. Mask (64 MiB) is
// L2-resident (192 MB), so halo re-reads are L2 hits. Vertical 31-tap box sum
// is done as a banded-matrix product on the WMMA pipe (f32 16x16x4, exact).

#define W       512
#define H       512
#define TILE_H  16
#define HROWS   48          // 46 live rows + 2 zero rows (K padded to 48)
#define LDS_STRIDE 520      // 512 + 8 to break LDS bank alignment across rows
#define KK_INV  (1.0f / 961.0f)
#define NBLOCKS 2048        // 64 images * 32 row-tiles

typedef __attribute__((ext_vector_type(2))) float v2f;
typedef __attribute__((ext_vector_type(8))) float v8f;

__global__ __launch_bounds__(256)
void sod_main(const float* __restrict__ pred, const float* __restrict__ mask,
              float* __restrict__ bceP, float* __restrict__ interP,
              float* __restrict__ unionP)
{
    __shared__ float hsum[HROWS * LDS_STRIDE];   // ~97.5 KB
    __shared__ float red[3 * 256];

    const int tid  = threadIdx.x;
    const int blk  = blockIdx.x;
    const int b    = blk >> 5;          // image index (32 tiles per image)
    const int tile = blk & 31;
    const int r0   = tile * TILE_H;

    const float* mimg = mask + (size_t)b * (H * W);
    const float* pimg = pred + (size_t)b * (H * W);

    // ---- Phase A: horizontal 31-tap zero-padded sliding sums into LDS ----
    // rows hr=0..45 <-> global rows r0-15 .. r0+30 ; hr=46,47 zeroed (K pad)
    for (int task = tid; task < HROWS * 16; task += 256) {
        const int hr  = task >> 4;
        const int seg = task & 15;
        const int cs  = seg * 32;
        const int gr  = r0 - 15 + hr;
        float* out = &hsum[hr * LDS_STRIDE + cs];
        if (hr >= 46 || gr < 0 || gr >= H) {
            #pragma unroll
            for (int i = 0; i < 32; ++i) out[i] = 0.0f;
        } else {
            const float* mrow = mimg + (size_t)gr * W;
            if (hr < TILE_H)  // overlap: prefetch the pred rows we'll stream later
                __builtin_prefetch(pimg + (size_t)(r0 + hr) * W + cs, 0, 0);
            float s = 0.0f;
            for (int x = cs - 15; x <= cs + 15; ++x)
                s += (x >= 0 && x < W) ? mrow[x] : 0.0f;
            for (int c = cs; c < cs + 32; ++c) {
                out[c - cs] = s;
                const float add = (c + 16 < W)  ? mrow[c + 16] : 0.0f;
                const float sub = (c - 15 >= 0) ? mrow[c - 15] : 0.0f;
                s += add - sub;
            }
        }
    }
    __syncthreads();

    // ---- Phase B: vertical 31-tap sum = band(16x48) x hsum(48x16) via WMMA,
    //      fused with weit/BCE/sigmoid/IoU accumulation -------------------
    const int lane = tid & 31;
    const int wave = tid >> 5;
    const int half = lane >> 4;
    const int nIdx = lane & 15;
    const int m    = nIdx;           // A-matrix row owned by this lane

    float bceAcc = 0.0f, interAcc = 0.0f, unAcc = 0.0f;

    for (int ctl = 0; ctl < 4; ++ctl) {
        const int c0 = (wave * 4 + ctl) * 16;   // wave owns cols [wave*64, +64)
        v8f acc = {};
        #pragma unroll
        for (int kk = 0; kk < 12; ++kk) {
            // A 16x4 layout: lanes 0-15 hold K = base+{0,1}, lanes 16-31 K = base+{2,3}
            const int kb = kk * 4 + 2 * half;
            v2f a, bf;
            a.x = (kb     >= m && kb     <= m + 30) ? 1.0f : 0.0f;
            a.y = (kb + 1 >= m && kb + 1 <= m + 30) ? 1.0f : 0.0f;
            bf.x = hsum[(kb)     * LDS_STRIDE + c0 + nIdx];
            bf.y = hsum[(kb + 1) * LDS_STRIDE + c0 + nIdx];
            acc = __builtin_amdgcn_wmma_f32_16x16x4_f32(
                false, a, false, bf, (short)0, acc, false, false);
        }
        // C/D layout: VGPR j -> M = j + 8*half, N = nIdx
        #pragma unroll
        for (int j = 0; j < 8; ++j) {
            const int row = r0 + j + 8 * half;
            const size_t idx = (size_t)row * W + c0 + nIdx;
            const float mv = mimg[idx];
            const float pv = pimg[idx];
            const float bm = acc[j] * KK_INV;                 // 31x31 box mean
            const float wt = 1.0f + 5.0f * fabsf(bm - mv);
            const float bce = fmaxf(pv, 0.0f) - pv * mv + log1pf(__expf(-fabsf(pv)));
            const float sg  = 1.0f / (1.0f + __expf(-pv));
            bceAcc   += bce;
            interAcc += sg * mv * wt;
            unAcc    += (sg + mv) * wt;
        }
    }

    // ---- block reduction (fixed order -> deterministic) ----
    __syncthreads();
    red[tid] = bceAcc; red[256 + tid] = interAcc; red[512 + tid] = unAcc;
    __syncthreads();
    for (int s = 128; s > 0; s >>= 1) {
        if (tid < s) {
            red[tid]       += red[tid + s];
            red[256 + tid] += red[256 + tid + s];
            red[512 + tid] += red[512 + tid + s];
        }
        __syncthreads();
    }
    if (tid == 0) {
        bceP[blk]   = red[0];
        interP[blk] = red[256];
        unionP[blk] = red[512];
    }
}

__global__ __launch_bounds__(256)
void sod_finalize(const float* __restrict__ ws, float* __restrict__ out)
{
    __shared__ float sb[256];
    __shared__ float sw[64];
    const int t = threadIdx.x;

    // total BCE sum over 2048 partials
    float s = 0.0f;
    for (int i = t; i < NBLOCKS; i += 256) s += ws[i];
    sb[t] = s;
    __syncthreads();
    for (int st = 128; st > 0; st >>= 1) {
        if (t < st) sb[t] += sb[t + st];
        __syncthreads();
    }

    // per-image wIoU from 32 tile partials each
    if (t < 64) {
        float inter = 0.0f, uni = 0.0f;
        for (int j = 0; j < 32; ++j) {
            inter += ws[NBLOCKS     + t * 32 + j];
            uni   += ws[2 * NBLOCKS + t * 32 + j];
        }
        sw[t] = 1.0f - (inter + 1.0f) / (uni - inter + 1.0f);
    }
    __syncthreads();

    if (t == 0) {
        float wsum = 0.0f;
        for (int j = 0; j < 64; ++j) wsum += sw[j];
        const float wbce = sb[0] / 16777216.0f;   // mean over 64*512*512
        out[0] = wbce + wsum / 64.0f;
    }
}

extern "C" void kernel_launch(void* const* d_in, const int* in_sizes, int n_in,
                              void* d_out, int out_size, void* d_ws, size_t ws_size,
                              hipStream_t stream)
{
    const float* pred = (const float*)d_in[0];
    const float* mask = (const float*)d_in[1];
    float* ws = (float*)d_ws;                 // needs 3*2048 floats = 24 KB
    float* bceP   = ws;
    float* interP = ws + NBLOCKS;
    float* unionP = ws + 2 * NBLOCKS;

    sod_main<<<dim3(NBLOCKS), dim3(256), 0, stream>>>(pred, mask, bceP, interP, unionP);
    sod_finalize<<<dim3(1), dim3(256), 0, stream>>>(ws, (float*)d_out);
}